// ZeroShotModule_84507776516742
// MI455X (gfx1250) — compile-verified
//
#include <hip/hip_runtime.h>
#include <hip/hip_bf16.h>
#include <math.h>

// ---------------------------------------------------------------------------
// ZeroShot entity-linking pipeline for gfx1250 (MI455X).
// All GEMMs go through V_WMMA_F32_16X16X4_F32 (native f32 tensor path).
// Inner loops are branchless with pointer-bumped addressing; out-of-range
// A rows / B cols are address-clamped (their garbage only reaches C rows/cols
// that are never stored).
// ---------------------------------------------------------------------------

typedef __attribute__((ext_vector_type(2))) float v2f;
typedef __attribute__((ext_vector_type(8))) float v8f;

#define PDIM 300     // mention/desc projection dim
#define EDIM 1000    // number of entity types

// ---------------------------------------------------------------------------
// f32 WMMA GEMM: C[M,N] = op(A[M,K] @ B[K,N] + bias), row-major.
// 128 threads = 4 waves; wave w computes a 32x16 tile at
//   (blockIdx.x*32, (blockIdx.y*4+w)*16): two accumulators share each B frag.
// K must be a multiple of 4; A rows must be 8B-aligned (K even).
// ---------------------------------------------------------------------------
__global__ __launch_bounds__(128)
void gemm_wmma_f32(const float* __restrict__ A, const float* __restrict__ B,
                   float* __restrict__ C, int M, int N, int K,
                   const float* __restrict__ bias, int relu)
{
    const int lane = threadIdx.x & 31;
    const int wave = threadIdx.x >> 5;
    const int half = lane >> 4;      // K sub-pair selector (ISA A-frag layout)
    const int mr   = lane & 15;

    const int tm = blockIdx.x * 32;
    const int tn = (blockIdx.y * 4 + wave) * 16;

    // clamped addresses: garbage from clamped rows/cols is never stored
    const int r0 = min(tm + mr,      M - 1);
    const int r1 = min(tm + 16 + mr, M - 1);
    const int cb = min(tn + mr,      N - 1);

    // loop-invariant per-lane bases; bumped by constants each K-step
    const float* ap0 = A + (size_t)r0 * K + 2 * half;
    const float* ap1 = A + (size_t)r1 * K + 2 * half;
    const float* bp0 = B + cb + (size_t)(2 * half) * N;
    const float* bp1 = bp0 + N;
    const size_t bstep = (size_t)4 * N;

    v8f acc0 = {}, acc1 = {};
    for (int k0 = 0; k0 < K; k0 += 4) {
        const v2f a0 = *(const v2f*)ap0;
        const v2f a1 = *(const v2f*)ap1;
        v2f b;
        b.x = *bp0;
        b.y = *bp1;
        // speculative gfx1250 global_prefetch_b8 ahead on the K streams
        __builtin_prefetch(ap0 + 64, 0, 1);
        __builtin_prefetch(ap1 + 64, 0, 1);
        __builtin_prefetch(bp0 + (size_t)16 * bstep, 0, 1);
        acc0 = __builtin_amdgcn_wmma_f32_16x16x4_f32(
            false, a0, false, b, (short)0, acc0, false, false);
        acc1 = __builtin_amdgcn_wmma_f32_16x16x4_f32(
            false, a1, false, b, (short)0, acc1, false, false);
        ap0 += 4; ap1 += 4; bp0 += bstep; bp1 += bstep;
    }

    const int colB = tn + mr;
    if (colB < N) {
        const float bv = bias ? bias[colB] : 0.0f;
        #pragma unroll
        for (int j = 0; j < 8; ++j) {
            int r = tm + j + 8 * half;         // ISA C/D layout
            if (r < M) {
                float v = acc0[j] + bv;
                if (relu) v = fmaxf(v, 0.0f);
                C[(size_t)r * N + colB] = v;
            }
            r += 16;
            if (r < M) {
                float v = acc1[j] + bv;
                if (relu) v = fmaxf(v, 0.0f);
                C[(size_t)r * N + colB] = v;
            }
        }
    }
}

// ---------------------------------------------------------------------------
// bio head: logits = log_softmax(h1 @ w2 + b2), pred = argmax. K=3 is tiny.
// ---------------------------------------------------------------------------
__global__ __launch_bounds__(256)
void bio_head_kernel(const float* __restrict__ h1, const float* __restrict__ w2,
                     const float* __restrict__ b2, float* __restrict__ logits,
                     int* __restrict__ pred, int Nrows, int H)
{
    const int r = blockIdx.x * blockDim.x + threadIdx.x;
    if (r >= Nrows) return;
    float s0 = b2[0], s1 = b2[1], s2 = b2[2];
    const float* hr = h1 + (size_t)r * H;
    for (int k = 0; k < H; ++k) {
        const float h = hr[k];
        s0 += h * w2[k * 3 + 0];
        s1 += h * w2[k * 3 + 1];
        s2 += h * w2[k * 3 + 2];
    }
    const float mx  = fmaxf(s0, fmaxf(s1, s2));
    const float lse = mx + __logf(__expf(s0 - mx) + __expf(s1 - mx) + __expf(s2 - mx));
    logits[(size_t)r * 3 + 0] = s0 - lse;
    logits[(size_t)r * 3 + 1] = s1 - lse;
    logits[(size_t)r * 3 + 2] = s2 - lse;
    int am = 0; float bv = s0;
    if (s1 > bv) { bv = s1; am = 1; }
    if (s2 > bv) { bv = s2; am = 2; }
    pred[r] = am;
}

// ---------------------------------------------------------------------------
// zero fill
// ---------------------------------------------------------------------------
__global__ __launch_bounds__(256)
void zero_f32(float* __restrict__ p, size_t n)
{
    size_t i = (size_t)blockIdx.x * blockDim.x + threadIdx.x;
    const size_t stride = (size_t)gridDim.x * blockDim.x;
    for (; i < n; i += stride) p[i] = 0.0f;
}

// ---------------------------------------------------------------------------
// Sequential BIO segment scan (inherently serial; 32K iterations is trivial).
// ---------------------------------------------------------------------------
__global__ void scan_labels_kernel(const int* __restrict__ labels,
                                   int* __restrict__ seg, float* __restrict__ cnt,
                                   int* __restrict__ c_row, int* __restrict__ starts,
                                   int Bn, int Ln)
{
    if (threadIdx.x != 0 || blockIdx.x != 0) return;
    const int N = Bn * Ln;
    int cum = 0;
    for (int n = 0; n < N; ++n) {
        const int lab = labels[n];
        if (lab == 1) cum++;
        const int s = (lab != 0 && cum > 0) ? (cum - 1) : -1;
        seg[n] = s;
        if (s >= 0) cnt[s] += 1.0f;
    }
    int run = 0;
    for (int b = 0; b < Bn; ++b) {
        int cc = 0;
        for (int l = 0; l < Ln; ++l)
            if (labels[b * Ln + l] == 1) cc++;
        c_row[b] = cc;
        starts[b] = run;
        run += cc;
    }
}

// ---------------------------------------------------------------------------
// Per-token accumulation into segment sums (f32 global atomics) + means.
// ---------------------------------------------------------------------------
__global__ __launch_bounds__(256)
void seg_accum_kernel(const int* __restrict__ seg, const float* __restrict__ hidden,
                      float* __restrict__ sums, int H)
{
    const int n = blockIdx.x;
    const int s = seg[n];
    if (s < 0) return;
    const float* hr = hidden + (size_t)n * H;
    float* sr = sums + (size_t)s * H;
    for (int h = threadIdx.x; h < H; h += blockDim.x)
        atomicAdd(&sr[h], hr[h]);
}

__global__ __launch_bounds__(256)
void seg_mean_kernel(float* __restrict__ sums, const float* __restrict__ cnt,
                     int S, int H)
{
    const size_t i = (size_t)blockIdx.x * blockDim.x + threadIdx.x;
    if (i >= (size_t)S * H) return;
    const int s = (int)(i / (size_t)H);
    sums[i] /= fmaxf(cnt[s], 1.0f);
}

// ---------------------------------------------------------------------------
// Fused gather + [16,300]x[300,1000] WMMA + log-softmax(E=1000).
// One block = 16 consecutive (b,l) rows; 8 waves stride 32 entity tile-PAIRS
// (each wave iteration: one shared A frag feeds two 16-entity tiles).
// Score matrix [16,1000] lives in 64000B of LDS; never touches HBM.
// Padded rows point at a zero row -> scores 0 -> output -log(E), as reference.
// ---------------------------------------------------------------------------
__global__ __launch_bounds__(256)
void mention_entity_logits(const float* __restrict__ seg_proj,
                           const float* __restrict__ proj_desc,
                           const float* __restrict__ zrow,   // PDIM zeros
                           const int* __restrict__ c_row,
                           const int* __restrict__ starts,
                           float* __restrict__ out, int Ln)
{
    __shared__ float Sc[16 * EDIM];   // 64000 bytes

    const int t    = threadIdx.x;
    const int lane = t & 31;
    const int wave = t >> 5;          // 0..7
    const int half = lane >> 4;
    const int mr   = lane & 15;
    const int rowBase = blockIdx.x * 16;

    // A-fragment row gather: this lane supplies mention row (rowBase+mr)
    const int flatA = rowBase + mr;
    const int bA = flatA / Ln, lA = flatA % Ln;
    const float* arowBase = (lA < c_row[bA])
        ? (seg_proj + (size_t)(starts[bA] + lA) * PDIM)
        : zrow;

    for (int et2 = wave; et2 < 32; et2 += 8) {   // 32 tile-pairs cover e<1024
        const int e0 = et2 * 32 + mr;            // B-frag n = mr
        const int e1 = e0 + 16;
        const float* ap  = arowBase + 2 * half;
        const float* bq0 = proj_desc + (size_t)min(e0, EDIM - 1) * PDIM + 2 * half;
        const float* bq1 = proj_desc + (size_t)min(e1, EDIM - 1) * PDIM + 2 * half;
        v8f acc0 = {}, acc1 = {};
        for (int k0 = 0; k0 < PDIM; k0 += 4) {   // 75 K-steps
            const v2f a  = *(const v2f*)ap;
            const v2f b0 = *(const v2f*)bq0;
            const v2f b1 = *(const v2f*)bq1;
            acc0 = __builtin_amdgcn_wmma_f32_16x16x4_f32(
                false, a, false, b0, (short)0, acc0, false, false);
            acc1 = __builtin_amdgcn_wmma_f32_16x16x4_f32(
                false, a, false, b1, (short)0, acc1, false, false);
            ap += 4; bq0 += 4; bq1 += 4;
        }
        #pragma unroll
        for (int j = 0; j < 8; ++j) {
            const int r = j + 8 * half;          // ISA C/D layout
            if (e0 < EDIM) Sc[r * EDIM + e0] = acc0[j];
            if (e1 < EDIM) Sc[r * EDIM + e1] = acc1[j];
        }
    }
    __syncthreads();

    // log-softmax over the E axis; one wave per row (rows strided by 8)
    for (int r = wave; r < 16; r += 8) {
        float mx = -3.402823466e38f;
        for (int e = lane; e < EDIM; e += 32) mx = fmaxf(mx, Sc[r * EDIM + e]);
        for (int o = 16; o > 0; o >>= 1) mx = fmaxf(mx, __shfl_xor(mx, o, 32));
        float sum = 0.0f;
        for (int e = lane; e < EDIM; e += 32) sum += __expf(Sc[r * EDIM + e] - mx);
        for (int o = 16; o > 0; o >>= 1) sum += __shfl_xor(sum, o, 32);
        const float lz = mx + __logf(sum);
        float* orow = out + (size_t)(rowBase + r) * EDIM;
        for (int e = lane; e < EDIM; e += 32) orow[e] = Sc[r * EDIM + e] - lz;
    }
}

// ---------------------------------------------------------------------------
// host driver
// ---------------------------------------------------------------------------
extern "C" void kernel_launch(void* const* d_in, const int* in_sizes, int n_in,
                              void* d_out, int out_size, void* d_ws, size_t ws_size,
                              hipStream_t stream)
{
    const int*   labels    = (const int*)  d_in[0];  // [B,L]
    const float* hidden    = (const float*)d_in[1];  // [B,L,H]
    const float* ety       = (const float*)d_in[2];  // [E,H]
    const float* w1        = (const float*)d_in[3];  // [H,H]
    const float* b1        = (const float*)d_in[4];  // [H]
    const float* w2        = (const float*)d_in[5];  // [H,3]
    const float* b2        = (const float*)d_in[6];  // [3]
    const float* w_mention = (const float*)d_in[7];  // [H,P]
    const float* w_desc    = (const float*)d_in[8];  // [H,P]

    const int H    = in_sizes[4];        // 768
    const int Ntok = in_sizes[0];        // B*L = 32768
    const int E    = in_sizes[2] / H;    // 1000
    const int P    = in_sizes[7] / H;    // 300
    const int Lrow = 512;                // sequence length (reference)
    const int Brow = Ntok / Lrow;        // 64

    // --- workspace carve ---
    char* ws = (char*)d_ws;
    size_t off = 0;
    auto carve = [&](size_t bytes) -> char* {
        char* p = ws + off;
        off = (off + bytes + 255) & ~(size_t)255;
        return p;
    };
    float* proj_desc = (float*)carve((size_t)E * P * 4);       // [E,P]
    float* h1sums    = (float*)carve((size_t)Ntok * H * 4);    // h1, then seg sums/means
    int*   pred      = (int*)  carve((size_t)Ntok * 4);
    int*   seg       = (int*)  carve((size_t)Ntok * 4);
    float* cnt       = (float*)carve((size_t)Ntok * 4);
    int*   c_row     = (int*)  carve((size_t)Brow * 4);
    int*   starts    = (int*)  carve((size_t)Brow * 4);
    float* zrow      = (float*)carve((size_t)P * 4);           // zero mention row
    float* seg_proj  = (float*)carve((size_t)Ntok * P * 4);    // [S,P]
    (void)ws_size; (void)n_in; (void)out_size;

    float* out_logits = (float*)d_out;                  // [B,L,3]
    float* out_s1 = out_logits + (size_t)Ntok * 3;      // [B,L,E]
    float* out_s2 = out_s1 + (size_t)Ntok * E;          // [B,L,E]

    const dim3 gblk(128);

    zero_f32<<<1, 256, 0, stream>>>(zrow, (size_t)P);

    // proj_desc = entity_type_embeddings @ w_desc  [E,H]x[H,P]
    gemm_wmma_f32<<<dim3((E + 31) / 32, (P + 63) / 64), gblk, 0, stream>>>(
        ety, w_desc, proj_desc, E, P, H, nullptr, 0);

    // h1 = relu(hidden @ w1 + b1)  [N,H]x[H,H]
    gemm_wmma_f32<<<dim3((Ntok + 31) / 32, (H + 63) / 64), gblk, 0, stream>>>(
        hidden, w1, h1sums, Ntok, H, H, b1, 1);

    // logits + argmax
    bio_head_kernel<<<(Ntok + 255) / 256, 256, 0, stream>>>(
        h1sums, w2, b2, out_logits, pred, Ntok, H);

    for (int v = 0; v < 2; ++v) {
        const int* lab = (v == 0) ? labels : pred;
        float* outv    = (v == 0) ? out_s1 : out_s2;

        zero_f32<<<4096, 256, 0, stream>>>(h1sums, (size_t)Ntok * H);
        zero_f32<<<64, 256, 0, stream>>>(cnt, (size_t)Ntok);
        scan_labels_kernel<<<1, 1, 0, stream>>>(lab, seg, cnt, c_row, starts, Brow, Lrow);
        seg_accum_kernel<<<Ntok, 256, 0, stream>>>(seg, hidden, h1sums, H);
        seg_mean_kernel<<<(unsigned)(((size_t)Ntok * H + 255) / 256), 256, 0, stream>>>(
            h1sums, cnt, Ntok, H);

        // seg_proj = segment_means @ w_mention  [S,H]x[H,P]
        gemm_wmma_f32<<<dim3((Ntok + 31) / 32, (P + 63) / 64), gblk, 0, stream>>>(
            h1sums, w_mention, seg_proj, Ntok, P, H, nullptr, 0);

        // fused gather + score GEMM + log-softmax
        mention_entity_logits<<<Ntok / 16, 256, 0, stream>>>(
            seg_proj, proj_desc, zrow, c_row, starts, outv, Lrow);
    }
}